// HorizontalDilation_10514079941475
// MI455X (gfx1250) — compile-verified
//
#include <hip/hip_runtime.h>
#include <math.h>

// HorizontalDilation(size=7): out = max(0, sliding-horizontal-max(img, width 15))
// Memory-bound: 128 MiB @ 23.3 TB/s ~ 5.6 us floor. CDNA5 path: async
// global->LDS staging (ASYNCcnt), aligned ds_load_b128 window reads,
// shared-window max factorization (~6 v_max_f32 per output), NT b128 stores.

#define IMG_W 4096
#define IMG_H 4096
#define TILE  1024                // pixels per block (4 per thread, 256 threads)
#define HALO  16                  // padded halo (>= 7), keeps 16B chunk alignment
#define LDSN  (TILE + 2 * HALO)   // 1056 floats = 4224 B LDS
#define NCHUNK (LDSN / 4)         // 264 x b128 chunks

typedef float v4f __attribute__((ext_vector_type(4)));

__global__ __launch_bounds__(256)
void HorizontalDilation_10514079941475_kernel(const float* __restrict__ img,
                                              float* __restrict__ out) {
    __shared__ float s[LDSN];

    const int tid  = threadIdx.x;
    const int bid  = blockIdx.x;
    const int row  = bid >> 2;                 // 4 tiles per row
    const int tile = bid & 3;
    const int c0   = tile * TILE;
    const float* rowp = img + (size_t)row * IMG_W;

    // ---- Stage tile + halo into LDS with CDNA5 async global->LDS copies ----
    // Chunk ch covers row-local floats [c0 - HALO + 4*ch, +4). Out-of-row
    // chunks exist only at tile 0 (first 4) and tile 3 (last 4) and are
    // handled by a block-uniform -INF fill, so the async loop is branch-free.
    const int chStart = (tile == 0) ? 4 : 0;
    const int chEnd   = (tile == 3) ? (NCHUNK - 4) : NCHUNK;
    for (int ch = chStart + tid; ch < chEnd; ch += 256) {
        unsigned lofs   = (unsigned)(size_t)(const void*)&s[4 * ch]; // LDS byte addr
        const float* gp = rowp + (c0 - HALO + 4 * ch);               // 16B-aligned
        asm volatile("global_load_async_to_lds_b128 %0, %1, off"
                     :
                     : "v"(lofs), "v"(gp)
                     : "memory");
    }
    if (tile == 0 && tid < 16) {               // fill s[0..15] = -inf
        s[tid] = -INFINITY;
    }
    if (tile == 3 && tid < 16) {               // fill s[1040..1055] = -inf
        s[LDSN - 16 + tid] = -INFINITY;
    }

    // Each wave drains its own ASYNCcnt; barrier publishes all LDS writes.
    asm volatile("s_wait_asynccnt 0" ::: "memory");
    __syncthreads();

    // ---- Compute 4 outputs per thread from 5 aligned ds_load_b128 reads ----
    // a[j] = s[8 + 4*tid + j], j = 0..19 (16B-aligned base). Output x+i
    // (x = c0 + 4*tid) covers a[1+i .. 15+i]; intersection a[4..15].
    const int b = 8 + 4 * tid;
    float a[20];
    const v4f* sv = (const v4f*)&s[b];
#pragma unroll
    for (int q = 0; q < 5; ++q) {
        v4f v = sv[q];
        a[4 * q + 0] = v.x;
        a[4 * q + 1] = v.y;
        a[4 * q + 2] = v.z;
        a[4 * q + 3] = v.w;
    }

    float common = a[4];
#pragma unroll
    for (int j = 5; j <= 15; ++j) common = fmaxf(common, a[j]);

    const float L2 = fmaxf(a[2], a[3]);        // left prefix pair
    const float R2 = fmaxf(a[16], a[17]);      // right suffix pair

    float o0 = fmaxf(common, fmaxf(a[1], L2));     // a[1..15]
    float o1 = fmaxf(common, fmaxf(L2, a[16]));    // a[2..16]
    float o2 = fmaxf(common, fmaxf(a[3], R2));     // a[3..17]
    float o3 = fmaxf(common, fmaxf(R2, a[18]));    // a[4..18]

    o0 = fmaxf(o0, 0.0f);
    o1 = fmaxf(o1, 0.0f);
    o2 = fmaxf(o2, 0.0f);
    o3 = fmaxf(o3, 0.0f);

    // 16B-aligned streaming store -> global_store_b128 (non-temporal hint)
    v4f r = {o0, o1, o2, o3};
    v4f* op = (v4f*)(out + (size_t)row * IMG_W + c0 + 4 * tid);
    __builtin_nontemporal_store(r, op);
}

extern "C" void kernel_launch(void* const* d_in, const int* in_sizes, int n_in,
                              void* d_out, int out_size, void* d_ws, size_t ws_size,
                              hipStream_t stream) {
    const float* img = (const float*)d_in[0];
    float* out = (float*)d_out;
    const int blocks = IMG_H * (IMG_W / TILE);   // 16384 blocks x 256 threads (8 wave32)
    HorizontalDilation_10514079941475_kernel<<<blocks, 256, 0, stream>>>(img, out);
}